// RelativeAttentionPixelClassifier_76802605187429
// MI455X (gfx1250) — compile-verified
//
#include <hip/hip_runtime.h>
#include <hip/hip_bf16.h>

typedef float v2f __attribute__((ext_vector_type(2)));
typedef float v8f __attribute__((ext_vector_type(8)));

#define ES     33
#define DIMC   32
#define NH     4
#define HIMG   384
#define WIMG   384
#define HW     (HIMG * WIMG)
#define NCH    32
#define BIAS_N 256

// ---------------------------------------------------------------------------
// Kernel 1: fold q-projection, output head weights and scale into v[c][e]:
//   v[c,e] = sum_{hd} (w_out[hd/8]/sqrt(8)) * (seed_emb[c,:]@Wq[hd,:]) * Wk[hd,e]
// ---------------------------------------------------------------------------
__global__ __launch_bounds__(256) void k_prep(const float* __restrict__ seed_emb,
                                              const float* __restrict__ Wq,
                                              const float* __restrict__ Wk,
                                              const float* __restrict__ w_out,
                                              float* __restrict__ v) {
    __shared__ float u[NCH * 32];
    const float invscale = 0.35355339059327379f; // 1/sqrt(8)
    const int t = threadIdx.x;
    for (int idx = t; idx < NCH * 32; idx += 256) {
        const int c = idx >> 5, hd = idx & 31;
        float q = 0.f;
        for (int j = 0; j < DIMC; ++j)
            q += seed_emb[c * DIMC + j] * Wq[hd * DIMC + j];
        u[idx] = q * w_out[hd >> 3] * invscale;
    }
    __syncthreads();
    for (int idx = t; idx < NCH * ES; idx += 256) {
        const int c = idx / ES, e = idx % ES;
        float s = 0.f;
        for (int hd = 0; hd < 32; ++hd)
            s += u[c * 32 + hd] * Wk[hd * ES + e];
        v[idx] = s; // idx == c*ES + e
    }
}

// ---------------------------------------------------------------------------
// Kernel 2: biasterm[p] = b_out + sum_h w_out[h] * bilinear_256to384(bias_h)(p)
// (align_corners=True semantics)
// ---------------------------------------------------------------------------
__global__ __launch_bounds__(256) void k_bias(const float* __restrict__ rb,
                                              const float* __restrict__ w_out,
                                              const float* __restrict__ b_out,
                                              float* __restrict__ biasterm) {
    const int p = blockIdx.x * 256 + threadIdx.x;
    if (p >= HW) return;
    const int yo = p / WIMG, xo = p % WIMG;
    const float sc = (float)(BIAS_N - 1) / (float)(HIMG - 1);
    const float fy = yo * sc;
    const int y0 = (int)fy;
    const float wy = fy - (float)y0;
    const int y1 = min(y0 + 1, BIAS_N - 1);
    const float fx = xo * sc;
    const int x0 = (int)fx;
    const float wx = fx - (float)x0;
    const int x1 = min(x0 + 1, BIAS_N - 1);

    float acc = b_out[0];
#pragma unroll
    for (int h = 0; h < NH; ++h) {
        const float* bh = rb + (size_t)h * BIAS_N * BIAS_N;
        const float t00 = bh[y0 * BIAS_N + x0], t01 = bh[y0 * BIAS_N + x1];
        const float t10 = bh[y1 * BIAS_N + x0], t11 = bh[y1 * BIAS_N + x1];
        const float top = t00 * (1.f - wx) + t01 * wx;
        const float bot = t10 * (1.f - wx) + t11 * wx;
        acc += w_out[h] * (top * (1.f - wy) + bot * wy);
    }
    biasterm[p] = acc;
}

// ---------------------------------------------------------------------------
// Kernel 3: out[c,p] = v[c,:] . x[c,:,p] + biasterm[p]
// Streaming GEMV via V_WMMA_F32_16X16X4_F32:
//   A (16x4)  : 16 pixels x 4 embedding dims (per ISA layout:
//               VGPR0 lanes0-15 K=0 / lanes16-31 K=2; VGPR1 K=1 / K=3)
//   B (4x16)  : v[c, e0..e0+3] replicated over all 16 columns
//   C/D 16x16 : every column holds the dot products for the 16 pixels.
// K=33 -> 8 full chunks of 4 + one tail chunk (e=32, rest zero-padded).
// ---------------------------------------------------------------------------
__global__ __launch_bounds__(256) void k_main(const float* __restrict__ x,
                                              const float* __restrict__ v,
                                              const float* __restrict__ biasterm,
                                              float* __restrict__ out) {
    const int lane  = threadIdx.x & 31;
    const int wave  = threadIdx.x >> 5;
    const int c     = blockIdx.x / 36;
    const int blkInC = blockIdx.x % 36;
    const int m     = lane & 15;
    const int khalf = lane >> 4; // 0: K pair (0,1), 1: K pair (2,3)

    // B operands (wave-invariant per channel), per ISA 4x16 B layout
    const float* vc = v + c * ES;
    v2f bvec[9];
#pragma unroll
    for (int ch = 0; ch < 8; ++ch) {
        const int e0 = ch * 4 + khalf * 2;
        bvec[ch].x = vc[e0];
        bvec[ch].y = vc[e0 + 1];
    }
    bvec[8].x = (khalf == 0) ? vc[32] : 0.f; // tail: only e=32 valid
    bvec[8].y = 0.f;

    const float* xc = x + (size_t)c * (size_t)(ES * HW);
    const int tile0 = blkInC * 256 + wave * 32; // 16-pixel tiles, per channel

    for (int t = 0; t < 32; ++t) {
        const int p0 = (tile0 + t) * 16;
        v8f acc = {};
#pragma unroll
        for (int ch = 0; ch < 8; ++ch) {
            const int e0 = ch * 4 + khalf * 2;
            v2f a;
            a.x = xc[(size_t)e0 * HW + p0 + m];
            a.y = xc[(size_t)(e0 + 1) * HW + p0 + m];
            acc = __builtin_amdgcn_wmma_f32_16x16x4_f32(
                false, a, false, bvec[ch], (short)0, acc, false, false);
        }
        { // tail chunk: K = {32, pad, pad, pad}
            v2f a;
            a.x = (khalf == 0) ? xc[(size_t)32 * HW + p0 + m] : 0.f;
            a.y = 0.f;
            acc = __builtin_amdgcn_wmma_f32_16x16x4_f32(
                false, a, false, bvec[8], (short)0, acc, false, false);
        }

        // D layout: VGPR r, lanes0-15 -> pixel p0+r ; lanes16-31 -> p0+8+r.
        // All 16 columns are identical, so select element (lane&7), then one
        // shuffle gives lane L (<16) the value for pixel p0+L.
        const int j = lane & 7;
        float sel = acc[0];
        sel = (j == 1) ? acc[1] : sel;
        sel = (j == 2) ? acc[2] : sel;
        sel = (j == 3) ? acc[3] : sel;
        sel = (j == 4) ? acc[4] : sel;
        sel = (j == 5) ? acc[5] : sel;
        sel = (j == 6) ? acc[6] : sel;
        sel = (j == 7) ? acc[7] : sel;
        const int l15 = lane & 15;
        const int src = (l15 < 8) ? l15 : (l15 + 8);
        const float r = __shfl(sel, src, 32);
        if (lane < 16) {
            const int p = p0 + lane;
            out[(size_t)c * HW + p] = r + biasterm[p];
        }
    }
}

// ---------------------------------------------------------------------------
extern "C" void kernel_launch(void* const* d_in, const int* in_sizes, int n_in,
                              void* d_out, int out_size, void* d_ws, size_t ws_size,
                              hipStream_t stream) {
    const float* x        = (const float*)d_in[0];
    const float* seed_emb = (const float*)d_in[1];
    const float* Wq       = (const float*)d_in[2];
    const float* Wk       = (const float*)d_in[3];
    const float* rb       = (const float*)d_in[4];
    const float* w_out    = (const float*)d_in[5];
    const float* b_out    = (const float*)d_in[6];
    float* out = (float*)d_out;

    float* ws_v    = (float*)d_ws;        // 32*33 floats
    float* ws_bias = (float*)d_ws + 2048; // HW floats

    k_prep<<<1, 256, 0, stream>>>(seed_emb, Wq, Wk, w_out, ws_v);
    k_bias<<<(HW + 255) / 256, 256, 0, stream>>>(rb, w_out, b_out, ws_bias);
    k_main<<<NCH * 36, 256, 0, stream>>>(x, ws_v, ws_bias, out);
}